// SINEModel_55224689492778
// MI455X (gfx1250) — compile-verified
//
#include <hip/hip_runtime.h>
#include <hip/hip_bf16.h>

// Problem constants (from reference)
#define BN 2048
#define SN 50
#define DN 64
#define LN 500
#define KN 4
#define NN 100000
#define LN_EPS 1e-12f
#define TAU 0.1f

#define TS 68  // LDS row stride (floats): bank = (4*row + k) % 64 -> conflict-free WMMA operand reads

typedef __attribute__((ext_vector_type(2))) float v2f;
typedef __attribute__((ext_vector_type(8))) float v8f;

#if defined(__has_builtin)
#if __has_builtin(__builtin_amdgcn_global_load_async_to_lds_b128)
#define HAVE_ASYNC_LDS 1
#endif
#endif

#ifdef HAVE_ASYNC_LDS
// Exact parameter types per hipcc diagnostic:
//   arg0: addrspace(1) pointer to 'int __attribute__((vector_size(16)))' (global src)
//   arg1: addrspace(3) pointer (LDS dst)
typedef int v4i __attribute__((vector_size(16)));
typedef __attribute__((address_space(1))) v4i* g_v4i_p;
typedef __attribute__((address_space(3))) v4i* l_v4i_p;
__device__ inline void async_copy16(const float* g, float* l) {
  float* gnc = const_cast<float*>(g);
  __builtin_amdgcn_global_load_async_to_lds_b128((g_v4i_p)gnc, (l_v4i_p)l, 0, 0);
}
__device__ inline void async_wait0() {
#if __has_builtin(__builtin_amdgcn_s_wait_asynccnt)
  __builtin_amdgcn_s_wait_asynccnt(0);
#else
  asm volatile("s_wait_asynccnt 0" ::: "memory");
#endif
}
#endif

// C[64x64] = A[64x64] @ B[64x64], all in LDS with row stride TS.
// 8 waves (256 threads); wave w computes output tiles w and w+8 of the 4x4 tile grid.
// V_WMMA_F32_16X16X4_F32: A 16x4 (lane<16: K=0/1, lane>=16: K=2/3), B 4x16 mirrored,
// C/D: VGPR j holds M=j (lanes 0-15) / M=j+8 (lanes 16-31), N=lane%16.
__device__ inline void gemm64(const float* __restrict__ A,
                              const float* __restrict__ Bm,
                              float* __restrict__ Cm) {
  const int lane  = threadIdx.x & 31;
  const int wave  = __builtin_amdgcn_readfirstlane(threadIdx.x >> 5);  // scalar -> uniform loop
  const int row   = lane & 15;          // M (for A) / N (for B)
  const int khalf = (lane >> 4) << 1;   // 0 or 2
  for (int t = wave; t < 16; t += 8) {
    const int mt = t >> 2, nt = t & 3;
    v8f acc = {};
    const float* Ab = A + (mt * 16 + row) * TS;
    const float* Bb = Bm + nt * 16 + row;
#pragma unroll
    for (int kk = 0; kk < 16; ++kk) {
      const int kb = kk * 4 + khalf;
      v2f a; a.x = Ab[kb];       a.y = Ab[kb + 1];
      v2f b; b.x = Bb[kb * TS];  b.y = Bb[(kb + 1) * TS];
      acc = __builtin_amdgcn_wmma_f32_16x16x4_f32(false, a, false, b,
                                                  (short)0, acc, false, false);
    }
    float* Cb = Cm + (mt * 16 + 8 * (lane >> 4)) * TS + nt * 16 + (lane & 15);
#pragma unroll
    for (int j = 0; j < 8; ++j) Cb[j * TS] = acc[j];
  }
}

// ---------------------------------------------------------------------------
// Kernel 1: one block (256 threads) per batch row; computes v_u[b][0..63].
// ---------------------------------------------------------------------------
__global__ __launch_bounds__(256) void sine_batch_kernel(
    const int* __restrict__ item_seq, const float* __restrict__ item_emb,
    const float* __restrict__ Cproto,
    const float* __restrict__ w1, const float* __restrict__ w2,
    const float* __restrict__ w3, const float* __restrict__ w4,
    const float* __restrict__ wk1, const float* __restrict__ wk2,
    const float* __restrict__ ln2w, const float* __restrict__ ln2b,
    const float* __restrict__ ln4w, const float* __restrict__ ln4b,
    float* __restrict__ vu) {
  __shared__ float s_xu[64 * TS];     // x_u padded to 64 rows (rows 50..63 zero); later x_u_bar
  __shared__ float s_w[64 * TS];      // current 64x64 weight
  __shared__ float s_tmp[64 * TS];    // gemm output
  __shared__ float s_w3x[SN * 64];    // l2norm(x_u @ w3)
  __shared__ float s_x[SN];           // per-seq scalars (reused)
  __shared__ float s_zu[64];
  __shared__ float s_su[LN];
  __shared__ int   s_idx[KN];
  __shared__ float s_val[KN];         // top-k scores, later e_k
  __shared__ float s_Cu[KN * 64];
  __shared__ float s_Cun[KN * 64];
  __shared__ float s_Pkt[SN * KN];    // P_k_t_b[s][k]
  __shared__ float s_Ptk[KN * SN];    // P_t_k, then mul_p
  __shared__ float s_delta[KN * 64];
  __shared__ float s_capt[64];
  __shared__ float s_red[8];

  const int b   = blockIdx.x;
  const int tid = threadIdx.x;

  // gather x_u rows (zero-pad rows 50..63) and load w1
  for (int i = tid; i < 64 * 64; i += 256) {
    const int s = i >> 6, d = i & 63;
    float v = 0.f;
    if (s < SN) v = item_emb[(size_t)item_seq[b * SN + s] * DN + d];
    s_xu[s * TS + d] = v;
  }
  for (int i = tid; i < 64 * 64; i += 256) s_w[(i >> 6) * TS + (i & 63)] = w1[i];
  __syncthreads();

  gemm64(s_xu, s_w, s_tmp);           // tanh-arg of stage-1 attention
  __syncthreads();

  if (tid < SN) {                     // x[s] = sum_d tanh(.)*w2[d]
    float acc = 0.f;
    for (int d = 0; d < 64; ++d) acc += tanhf(s_tmp[tid * TS + d]) * w2[d];
    s_x[tid] = acc;
  }
  __syncthreads();
  if (tid == 0) {                     // softmax over s
    float m = -1e30f;
    for (int s = 0; s < SN; ++s) m = fmaxf(m, s_x[s]);
    float sum = 0.f;
    for (int s = 0; s < SN; ++s) { const float e = __expf(s_x[s] - m); s_x[s] = e; sum += e; }
    const float r = 1.f / sum;
    for (int s = 0; s < SN; ++s) s_x[s] *= r;
  }
  __syncthreads();
  if (tid < 64) {                     // z_u
    float acc = 0.f;
    for (int s = 0; s < SN; ++s) acc += s_x[s] * s_xu[s * TS + tid];
    s_zu[tid] = acc;
  }
  __syncthreads();
  for (int l = tid; l < LN; l += 256) {   // s_u = z_u @ C^T
    float acc = 0.f;
    for (int d = 0; d < 64; ++d) acc += s_zu[d] * Cproto[l * DN + d];
    s_su[l] = acc;
  }
  __syncthreads();
  if (tid == 0) {                     // top-4, kept ascending
    int idx[KN]; float val[KN];
    for (int k = 0; k < KN; ++k) { idx[k] = 0; val[k] = -1e30f; }
    for (int l = 0; l < LN; ++l) {
      const float v = s_su[l];
      if (v > val[0]) {
        int p = 0;
        while (p < KN - 1 && v > val[p + 1]) { val[p] = val[p + 1]; idx[p] = idx[p + 1]; ++p; }
        val[p] = v; idx[p] = l;
      }
    }
    for (int k = 0; k < KN; ++k) { s_idx[k] = idx[k]; s_val[k] = val[k]; }
  }
  __syncthreads();
  {                                   // C_u = C[idx] * sigmoid(s_top)
    const int k = tid >> 6, d = tid & 63;
    const float sg = 1.f / (1.f + __expf(-s_val[k]));
    s_Cu[k * 64 + d] = Cproto[s_idx[k] * DN + d] * sg;
  }
  __syncthreads();
  if (tid < KN) {                     // layernorm(C_u) rows
    float m = 0.f;
    for (int d = 0; d < 64; ++d) m += s_Cu[tid * 64 + d];
    m *= (1.f / 64.f);
    float v = 0.f;
    for (int d = 0; d < 64; ++d) { const float t = s_Cu[tid * 64 + d] - m; v += t * t; }
    v *= (1.f / 64.f);
    const float r = rsqrtf(v + LN_EPS);
    for (int d = 0; d < 64; ++d)
      s_Cun[tid * 64 + d] = (s_Cu[tid * 64 + d] - m) * r * ln2w[d] + ln2b[d];
  }
  for (int i = tid; i < 64 * 64; i += 256) s_w[(i >> 6) * TS + (i & 63)] = w3[i];
  __syncthreads();

  gemm64(s_xu, s_w, s_tmp);           // x_u @ w3
  __syncthreads();
  if (tid < SN) {                     // l2norm rows -> s_w3x
    float n = 0.f;
    for (int d = 0; d < 64; ++d) { const float t = s_tmp[tid * TS + d]; n += t * t; }
    const float r = 1.f / fmaxf(sqrtf(n), 1e-12f);
    for (int d = 0; d < 64; ++d) s_w3x[tid * 64 + d] = s_tmp[tid * TS + d] * r;
  }
  __syncthreads();
  if (tid < SN) {                     // P_k_t + softmax over k
    float p[KN]; float m = -1e30f;
    for (int k = 0; k < KN; ++k) {
      float acc = 0.f;
      for (int d = 0; d < 64; ++d) acc += s_w3x[tid * 64 + d] * s_Cun[k * 64 + d];
      p[k] = acc; m = fmaxf(m, acc);
    }
    float sum = 0.f;
    for (int k = 0; k < KN; ++k) { p[k] = __expf(p[k] - m); sum += p[k]; }
    const float r = 1.f / sum;
    for (int k = 0; k < KN; ++k) s_Pkt[tid * KN + k] = p[k] * r;
  }
  // per-interest position attention: a_k = x_u @ w_k_1[k]
  for (int k = 0; k < KN; ++k) {
    __syncthreads();
    for (int i = tid; i < 64 * 64; i += 256)
      s_w[(i >> 6) * TS + (i & 63)] = wk1[k * 64 * 64 + i];
    __syncthreads();
    gemm64(s_xu, s_w, s_tmp);
    __syncthreads();
    if (tid < SN) {
      float acc = 0.f;
      for (int d = 0; d < 64; ++d) acc += tanhf(s_tmp[tid * TS + d]) * wk2[k * 64 + d];
      s_Ptk[k * SN + tid] = acc;
    }
  }
  __syncthreads();
  if (tid < KN) {                     // softmax over s, then fold in P_k_t_b -> mul_p
    float m = -1e30f;
    for (int s = 0; s < SN; ++s) m = fmaxf(m, s_Ptk[tid * SN + s]);
    float sum = 0.f;
    for (int s = 0; s < SN; ++s) { const float e = __expf(s_Ptk[tid * SN + s] - m); s_Ptk[tid * SN + s] = e; sum += e; }
    const float r = 1.f / sum;
    for (int s = 0; s < SN; ++s) s_Ptk[tid * SN + s] = s_Ptk[tid * SN + s] * r * s_Pkt[s * KN + tid];
  }
  __syncthreads();
  {                                   // delta_k = mul_p @ x_u   (256 threads = K*D exactly)
    const int k = tid >> 6, d = tid & 63;
    float acc = 0.f;
    for (int s = 0; s < SN; ++s) acc += s_Ptk[k * SN + s] * s_xu[s * TS + d];
    s_delta[k * 64 + d] = acc;
  }
  __syncthreads();
  if (tid < KN) {
    float n = 0.f;
    for (int d = 0; d < 64; ++d) { const float t = s_delta[tid * 64 + d]; n += t * t; }
    s_red[tid] = 1.f / fmaxf(sqrtf(n), 1e-12f);
  }
  __syncthreads();
  { const int k = tid >> 6, d = tid & 63; s_delta[k * 64 + d] *= s_red[k]; }
  __syncthreads();
  for (int i = tid; i < SN * 64; i += 256) {   // x_u_bar overwrites s_xu rows 0..49
    const int s = i >> 6, d = i & 63;
    float acc = 0.f;
    for (int k = 0; k < KN; ++k) acc += s_Pkt[s * KN + k] * s_Cu[k * 64 + d];
    s_xu[s * TS + d] = acc;                    // rows 50..63 remain zero
  }
  for (int i = tid; i < 64 * 64; i += 256) s_w[(i >> 6) * TS + (i & 63)] = w3[i];
  __syncthreads();

  gemm64(s_xu, s_w, s_tmp);           // x_u_bar @ w3
  __syncthreads();
  if (tid < SN) {
    float acc = 0.f;
    for (int d = 0; d < 64; ++d) acc += tanhf(s_tmp[tid * TS + d]) * w4[d];
    s_x[tid] = acc;
  }
  __syncthreads();
  if (tid == 0) {                     // softmax over s
    float m = -1e30f;
    for (int s = 0; s < SN; ++s) m = fmaxf(m, s_x[s]);
    float sum = 0.f;
    for (int s = 0; s < SN; ++s) { const float e = __expf(s_x[s] - m); s_x[s] = e; sum += e; }
    const float r = 1.f / sum;
    for (int s = 0; s < SN; ++s) s_x[s] *= r;
  }
  __syncthreads();
  if (tid < 64) {                     // C_apt vector
    float acc = 0.f;
    for (int s = 0; s < SN; ++s) acc += s_x[s] * s_xu[s * TS + tid];
    s_capt[tid] = acc;
  }
  __syncthreads();
  if (tid == 0) {                     // layernorm stats
    float m = 0.f;
    for (int d = 0; d < 64; ++d) m += s_capt[d];
    m *= (1.f / 64.f);
    float v = 0.f;
    for (int d = 0; d < 64; ++d) { const float t = s_capt[d] - m; v += t * t; }
    v *= (1.f / 64.f);
    s_red[0] = m; s_red[1] = rsqrtf(v + LN_EPS);
  }
  __syncthreads();
  if (tid < 64)
    s_capt[tid] = (s_capt[tid] - s_red[0]) * s_red[1] * ln4w[tid] + ln4b[tid];
  __syncthreads();
  if (tid < KN) {                     // e_k
    float acc = 0.f;
    for (int d = 0; d < 64; ++d) acc += s_delta[tid * 64 + d] * s_capt[d];
    s_val[tid] = acc * (1.f / TAU);
  }
  __syncthreads();
  if (tid == 0) {                     // softmax over k
    float m = -1e30f;
    for (int k = 0; k < KN; ++k) m = fmaxf(m, s_val[k]);
    float sum = 0.f;
    for (int k = 0; k < KN; ++k) { const float e = __expf(s_val[k] - m); s_val[k] = e; sum += e; }
    const float r = 1.f / sum;
    for (int k = 0; k < KN; ++k) s_val[k] *= r;
  }
  __syncthreads();
  if (tid < 64) {                     // v_u
    float acc = 0.f;
    for (int k = 0; k < KN; ++k) acc += s_val[k] * s_delta[k * 64 + tid];
    vu[(size_t)b * DN + tid] = acc;
  }
}

// ---------------------------------------------------------------------------
// Kernel 2: scores = v_u @ item_emb^T  (M=2048, N=100000, K=64), fp32 WMMA.
// Block = 128 threads (4 waves). Each block computes a 128-row x 64-col output
// macro-tile: B tile (64 item rows) staged ONCE into LDS (async-to-LDS DMA when
// available) and held in 32 VGPRs per wave across 8 M-tiles; A (vu) staged in
// LDS; output staged through LDS and written as fully coalesced float4 rows.
// item_emb (25.6 MB) and vu (512 KB) are L2-resident; the 819 MB of output
// stores set the runtime floor (~35 us at 23.3 TB/s).
// ---------------------------------------------------------------------------
__global__ __launch_bounds__(128) void sine_scores_kernel(
    const float* __restrict__ vu, const float* __restrict__ emb,
    float* __restrict__ out) {
  __shared__ float sA[128 * TS];   // 128 batch rows x 64   (~34.8 KB)
  __shared__ float sB[64 * TS];    // 64 item rows x 64     (~17.4 KB)
  __shared__ float sO[16 * 64];    // output staging tile   (4 KB)

  const int tid = threadIdx.x;
  const int m0  = blockIdx.y * 128;
  const int n0  = blockIdx.x * 64;

  // Stage B: thread t -> item row n0 + t/2, half-row (t&1)*32. Row start in LDS
  // is r*TS*4 = 272*r bytes (16B aligned), so b128 transfers are legal.
  {
    const int r  = tid >> 1;
    const int c0 = (tid & 1) * 32;
    if (n0 + r < NN) {
      const float* src = emb + (size_t)(n0 + r) * DN + c0;
      float*       dst = sB + r * TS + c0;
#ifdef HAVE_ASYNC_LDS
#pragma unroll
      for (int q = 0; q < 8; ++q) async_copy16(src + q * 4, dst + q * 4);
#else
#pragma unroll
      for (int q = 0; q < 8; ++q)
        *(float4*)(dst + q * 4) = *(const float4*)(src + q * 4);
#endif
    } else {
      for (int q = 0; q < 32; ++q) sB[r * TS + c0 + q] = 0.f;  // pad last N-block
    }
  }
  // Stage A: 128 rows x 64 of vu
  for (int i = tid; i < 128 * 64; i += 128) {
    const int r = i >> 6, c = i & 63;
    sA[r * TS + c] = vu[(size_t)(m0 + r) * DN + c];
  }
#ifdef HAVE_ASYNC_LDS
  async_wait0();
#endif
  __syncthreads();

  const int lane  = tid & 31;
  const int wave  = __builtin_amdgcn_readfirstlane(tid >> 5);  // 0..3, scalar
  const int n     = lane & 15;
  const int khalf = (lane >> 4) << 1;

  // Hoist this wave's B operands (16 cols x 64 k) into 32 VGPRs; reused 8x.
  v2f breg[16];
  {
    const float* Bb = sB + (wave * 16 + n) * TS;
#pragma unroll
    for (int kk = 0; kk < 16; ++kk) {
      const int kb = kk * 4 + khalf;
      breg[kk].x = Bb[kb]; breg[kk].y = Bb[kb + 1];
    }
  }

  for (int mt = 0; mt < 8; ++mt) {
    v8f acc = {};
    const float* Ab = sA + (mt * 16 + n) * TS;   // A row = M = lane%16
#pragma unroll
    for (int kk = 0; kk < 16; ++kk) {
      const int kb = kk * 4 + khalf;
      v2f a; a.x = Ab[kb]; a.y = Ab[kb + 1];
      acc = __builtin_amdgcn_wmma_f32_16x16x4_f32(false, a, false, breg[kk],
                                                  (short)0, acc, false, false);
    }
    // Stage accumulator tile to LDS (wave w owns cols w*16..w*16+15)
    float* Ob = sO + (8 * (lane >> 4)) * 64 + wave * 16 + n;
#pragma unroll
    for (int j = 0; j < 8; ++j) Ob[j * 64] = acc[j];
    __syncthreads();
    // Coalesced writeback: 16 rows x 64 cols, 8 floats (2 x b128) per thread
    {
      const int r  = tid >> 3;
      const int cb = (tid & 7) * 8;
      if (n0 + cb + 7 < NN) {   // NN % 8 == 0: whole 8-float group in or out
        const size_t gr = (size_t)(m0 + mt * 16 + r) * (size_t)NN + n0 + cb;
        *(float4*)(out + gr)     = *(const float4*)&sO[r * 64 + cb];
        *(float4*)(out + gr + 4) = *(const float4*)&sO[r * 64 + cb + 4];
      }
    }
    __syncthreads();
  }
}

extern "C" void kernel_launch(void* const* d_in, const int* in_sizes, int n_in,
                              void* d_out, int out_size, void* d_ws, size_t ws_size,
                              hipStream_t stream) {
  const int*   item_seq = (const int*)d_in[0];
  // d_in[1] = item_seq_len (unused by the reference forward)
  const float* item_emb = (const float*)d_in[2];
  const float* Cproto   = (const float*)d_in[3];
  const float* w1       = (const float*)d_in[4];
  const float* w2       = (const float*)d_in[5];
  const float* w3       = (const float*)d_in[6];
  const float* w4       = (const float*)d_in[7];
  const float* wk1      = (const float*)d_in[8];
  const float* wk2      = (const float*)d_in[9];
  const float* ln2w     = (const float*)d_in[10];
  const float* ln2b     = (const float*)d_in[11];
  const float* ln4w     = (const float*)d_in[12];
  const float* ln4b     = (const float*)d_in[13];
  float* scores = (float*)d_out;
  float* vu     = (float*)d_ws;       // B*D floats = 512 KB scratch

  sine_batch_kernel<<<BN, 256, 0, stream>>>(
      item_seq, item_emb, Cproto, w1, w2, w3, w4, wk1, wk2,
      ln2w, ln2b, ln4w, ln4b, vu);

  dim3 grid2((NN + 63) / 64, BN / 128);   // 1563 x 16
  sine_scores_kernel<<<grid2, 128, 0, stream>>>(vu, item_emb, scores);
}